// LinearAttention_14611478741383
// MI455X (gfx1250) — compile-verified
//
#include <hip/hip_runtime.h>
#include <hip/hip_bf16.h>

// Linear attention, fp32, gfx1250 (wave32):
//   Phase A: KV[n,h,d,v] = sum_s featK[s,d]*V[s,v]/S ; Ksum[d] = sum_s featK[s,d]
//            - K/V tiles streamed global->LDS with GLOBAL_LOAD_ASYNC_TO_LDS_B128
//              (ASYNCcnt, double-buffered), feature map applied once in LDS,
//              WMMA F32 16x16x4 operands fed from LDS.
//   Phase B: out[n,l,h,v] = (featQ . KV) * S / (featQ . Ksum + eps), LDS-staged GEMM.

#define N_ 8
#define L_ 4096
#define S_ 4096
#define H_ 8
#define D_ 64
#define EPS_ 1e-6f
#define SCHUNK 512
#define NCHUNKS (S_ / SCHUNK)
#define ST_ 32                    // s-rows per LDS stage (K tile 8KB + V tile 8KB)
#define NST (SCHUNK / ST_)

typedef __attribute__((ext_vector_type(2))) float v2f;
typedef __attribute__((ext_vector_type(4))) float v4f;
typedef __attribute__((ext_vector_type(8))) float v8f;

__device__ __forceinline__ float feat(float x) {
    // elu(x) + 1
    return x > 0.0f ? (x + 1.0f) : __expf(x);
}

// Low 32 bits of a flat shared-memory address are the LDS byte offset
// (aperture tag lives in addr[63:32], ISA 10.2).
__device__ __forceinline__ unsigned lds_off(const void* p) {
    return (unsigned)(unsigned long long)p;
}

// Per-lane 16B async copy global -> LDS, tracked by ASYNCcnt.
__device__ __forceinline__ void async_copy_b128(unsigned lds_byte, const float* gaddr) {
    asm volatile("global_load_async_to_lds_b128 %0, %1, off"
                 :: "v"(lds_byte), "v"(gaddr) : "memory");
}

__global__ void zero_ws_kernel(float* __restrict__ ws, int nfloats) {
    int i = blockIdx.x * blockDim.x + threadIdx.x;
    if (i < nfloats) ws[i] = 0.0f;
}

// ---------------- Phase A: KV = Kf^T @ (V/S), Ksum = colsum(Kf) ----------------
// grid (N*H, NCHUNKS), 512 threads = 16 waves; wave (tr,tc) owns a 16x16 tile of
// the 64x64 KV output. 32-row K/V stages double-buffered through LDS.
__global__ __launch_bounds__(512) void kv_kernel(
        const float* __restrict__ K, const float* __restrict__ V,
        const float* __restrict__ kvm,
        float* __restrict__ KVws, float* __restrict__ KsumWs) {
    __shared__ __align__(16) float sK[2][ST_ * D_];
    __shared__ __align__(16) float sV[2][ST_ * D_];
    __shared__ float sM[2][ST_];

    const int nh = blockIdx.x;
    const int n  = nh / H_;
    const int h  = nh % H_;
    const int s0 = blockIdx.y * SCHUNK;

    const int t     = threadIdx.x;
    const int wave  = t >> 5;
    const int lane  = t & 31;
    const int laneM = lane & 15;
    const int hi    = lane >> 4;      // lane half -> WMMA K offsets {0,1} vs {2,3}
    const int kk    = hi * 2;
    const int tr    = wave >> 2;
    const int tc    = wave & 3;
    const int d     = tr * 16 + laneM;   // KV row
    const int v     = tc * 16 + laneM;   // KV col
    const float scale = 1.0f / (float)S_;

    // staging: thread t copies 16B = 4 floats; row r, cols [c0,c0+4)
    const int r  = t >> 4;
    const int c0 = (t & 15) * 4;
    const int sstride = H_ * D_;
    const float* Kg = K + ((long)n * S_ * H_ + h) * D_;
    const float* Vg = V + ((long)n * S_ * H_ + h) * D_;
    const float* Mp = kvm + n * S_;

    // Prologue: stage 0 into buffer 0.
    async_copy_b128(lds_off(&sK[0][t * 4]), Kg + (size_t)(s0 + r) * sstride + c0);
    async_copy_b128(lds_off(&sV[0][t * 4]), Vg + (size_t)(s0 + r) * sstride + c0);
    if (t < ST_) sM[0][t] = Mp[s0 + t];

    v8f c = {0.f, 0.f, 0.f, 0.f, 0.f, 0.f, 0.f, 0.f};
    float ksumLocal = 0.0f;

    for (int st = 0; st < NST; ++st) {
        const int b = st & 1;

        if (st + 1 < NST) {
            // Kick next stage into the other buffer (freed by last iter's barrier).
            const int sb = s0 + (st + 1) * ST_;
            async_copy_b128(lds_off(&sK[b ^ 1][t * 4]), Kg + (size_t)(sb + r) * sstride + c0);
            async_copy_b128(lds_off(&sV[b ^ 1][t * 4]), Vg + (size_t)(sb + r) * sstride + c0);
            if (t < ST_) sM[b ^ 1][t] = Mp[sb + t];
            // Async loads complete in order per wave: <=2 outstanding means the
            // current stage's two copies have landed.
            asm volatile("s_wait_asynccnt 0x2" ::: "memory");
        } else {
            asm volatile("s_wait_asynccnt 0x0" ::: "memory");
        }
        __syncthreads();

        // In-place preprocess: Kf = feat(K)*mask (once per element), Vs = V*mask/S.
        {
            const float m  = sM[b][r];
            const float ms = m * scale;
            float* pk = &sK[b][t * 4];
            float* pv = &sV[b][t * 4];
#pragma unroll
            for (int i = 0; i < 4; ++i) pk[i] = feat(pk[i]) * m;
#pragma unroll
            for (int i = 0; i < 4; ++i) pv[i] *= ms;
        }
        __syncthreads();

        // 8 WMMA F32 16x16x4 over the 32 staged rows, operands from LDS.
#pragma unroll
        for (int ss = 0; ss < ST_; ss += 4) {
            v2f a, bb;
            a.x  = sK[b][(ss + kk) * D_ + d];
            a.y  = sK[b][(ss + kk + 1) * D_ + d];
            bb.x = sV[b][(ss + kk) * D_ + v];
            bb.y = sV[b][(ss + kk + 1) * D_ + v];
            c = __builtin_amdgcn_wmma_f32_16x16x4_f32(
                    false, a, false, bb, (short)0, c, false, false);
            ksumLocal += a.x + a.y;   // each (s,d) covered once per wave
        }
        __syncthreads();  // buffer b reusable for stage st+2
    }

    // Accumulate partial tile into workspace (8 chunk-blocks race via f32 atomics).
    float* KVdst = KVws + nh * (D_ * D_);
#pragma unroll
    for (int j = 0; j < 8; ++j) {
        const int row = tr * 16 + hi * 8 + j;   // C layout: VGPR j -> M=j / M=j+8
        atomicAdd(&KVdst[row * D_ + v], c[j]);
    }
    if (tc == 0) atomicAdd(&KsumWs[nh * D_ + d], ksumLocal);
}

// ---------------- Phase B: out = (Qf @ KV) * Z * S ----------------
// grid (N*H, L/64), 512 threads = 16 waves; KV + featured Q + Ksum + Z in LDS.
__global__ __launch_bounds__(512) void out_kernel(
        const float* __restrict__ Q, const float* __restrict__ qm,
        const float* __restrict__ KVws, const float* __restrict__ KsumWs,
        float* __restrict__ out) {
    __shared__ __align__(16) float sKV[D_ * D_];
    __shared__ __align__(16) float sQ[64 * D_];
    __shared__ float sKs[D_];
    __shared__ float sZ[64];

    const int nh = blockIdx.x;
    const int n  = nh / H_;
    const int h  = nh % H_;
    const int l0 = blockIdx.y * 64;
    const int t  = threadIdx.x;

    // Stage KV tile with b128 loads.
    {
        const v4f* KVv = (const v4f*)(KVws + nh * (D_ * D_));
        v4f* sKVv = (v4f*)sKV;
#pragma unroll
        for (int i = 0; i < 2; ++i) sKVv[t + i * 512] = KVv[t + i * 512];
    }
    if (t < D_) sKs[t] = KsumWs[nh * D_ + t];

    // Stage 64 featured Q rows with b128 loads (row-contiguous, 16B aligned).
#pragma unroll
    for (int i = 0; i < 2; ++i) {
        const int idx4 = t + i * 512;        // float4 index 0..1023
        const int flat = idx4 * 4;
        const int rr   = flat >> 6;
        const int dd   = flat & 63;
        const int l    = l0 + rr;
        const v4f q    = *(const v4f*)&Q[(((long)n * L_ + l) * H_ + h) * D_ + dd];
        const float qmv = qm[n * L_ + l];
        v4f o;
        o.x = feat(q.x) * qmv;
        o.y = feat(q.y) * qmv;
        o.z = feat(q.z) * qmv;
        o.w = feat(q.w) * qmv;
        *(v4f*)&sQ[flat] = o;
    }
    __syncthreads();

    // Z[l] = S / (Qf . Ksum + eps)   (the S folds away the 1/S baked into KV)
    if (t < 64) {
        float dot = 0.0f;
#pragma unroll
        for (int dd = 0; dd < D_; ++dd) dot += sQ[t * D_ + dd] * sKs[dd];
        sZ[t] = (float)S_ / (dot + EPS_);
    }
    __syncthreads();

    const int wave  = t >> 5;
    const int lane  = t & 31;
    const int laneM = lane & 15;
    const int hi    = lane >> 4;
    const int kk    = hi * 2;
    const int mr    = wave >> 2;          // 16-row tile
    const int nc    = wave & 3;           // 16-col tile
    const int col   = nc * 16 + laneM;
    const int rowA  = mr * 16 + laneM;

    v8f c = {0.f, 0.f, 0.f, 0.f, 0.f, 0.f, 0.f, 0.f};
#pragma unroll
    for (int k = 0; k < D_; k += 4) {
        v2f a, b;
        a.x = sQ[rowA * D_ + k + kk];          // A: Qf (16 l-rows x 4 d-cols)
        a.y = sQ[rowA * D_ + k + kk + 1];
        b.x = sKV[(k + kk) * D_ + col];        // B: KV (4 d-rows x 16 v-cols)
        b.y = sKV[(k + kk + 1) * D_ + col];
        c = __builtin_amdgcn_wmma_f32_16x16x4_f32(
                false, a, false, b, (short)0, c, false, false);
    }

#pragma unroll
    for (int j = 0; j < 8; ++j) {
        const int row = mr * 16 + hi * 8 + j;
        const int l   = l0 + row;
        out[(((long)n * L_ + l) * H_ + h) * D_ + col] = c[j] * sZ[row];
    }
}

extern "C" void kernel_launch(void* const* d_in, const int* in_sizes, int n_in,
                              void* d_out, int out_size, void* d_ws, size_t ws_size,
                              hipStream_t stream) {
    const float* Q      = (const float*)d_in[0];
    const float* K      = (const float*)d_in[1];
    const float* V      = (const float*)d_in[2];
    const float* qmask  = (const float*)d_in[3];
    const float* kvmask = (const float*)d_in[4];
    float* outp = (float*)d_out;

    float* KVws   = (float*)d_ws;                 // N*H*64*64 floats (1 MB)
    float* KsumWs = KVws + N_ * H_ * D_ * D_;     // N*H*64 floats
    const int nf  = N_ * H_ * D_ * D_ + N_ * H_ * D_;

    zero_ws_kernel<<<(nf + 255) / 256, 256, 0, stream>>>(KVws, nf);
    kv_kernel<<<dim3(N_ * H_, NCHUNKS), 512, 0, stream>>>(K, V, kvmask, KVws, KsumWs);
    out_kernel<<<dim3(N_ * H_, L_ / 64), 512, 0, stream>>>(Q, qmask, KVws, KsumWs, outp);
}